// MeGCN_30270929502752
// MI455X (gfx1250) — compile-verified
//
#include <hip/hip_runtime.h>

// MeGCN forward for MI455X (gfx1250, wave32).
// - fp32 WMMA (V_WMMA_F32_16X16X4_F32) for the two feature-projection GEMMs
// - A tile staged LDS-side via the Tensor Data Mover (tensor_load_to_lds +
//   s_wait_tensorcnt), with TDM padding for bank-conflict-free operand reads
// - W staged in WMMA-operand order so each B operand is one ds_load_b64
// - L2-resident COO SpMM via hardware f32 global atomics, modalities fused

#define N_USERS 100000
#define N_ITEMS 50000
#define N_TOT   150000
#define D_EMB   64
#define IMG_K   4096
#define TXT_K   384
#define ALPHA_F 0.8f
#define KCHUNK  64

#define B_STRIDE 160                              // floats per B pair-row (128 data + 32 pad)
#define A_STRIDE 98                               // floats per A row (64 data + 34 TDM pad)
#define LDSB_BYTES ((KCHUNK / 2) * B_STRIDE * 4)  // 32*160*4 = 20480
#define LDSA_BYTES (64 * A_STRIDE * 4)            // 25088
#define SMEM_BYTES (LDSB_BYTES + LDSA_BYTES)      // 45568 (< 64KB)

typedef __attribute__((ext_vector_type(2))) float        v2f;
typedef __attribute__((ext_vector_type(8))) float        v8f;
typedef __attribute__((ext_vector_type(4))) unsigned int v4u;
typedef __attribute__((ext_vector_type(4))) int          v4i;
typedef __attribute__((ext_vector_type(8))) int          v8i;

// ---------------------------------------------------------------------------
// GEMM  C[M,64] = A[M,K] @ W[K,64] + bias, then row-wise L2 normalization.
// Block = 128 threads (4 waves); each wave owns a 16x64 tile via 4 v8f
// accumulators.  K % KCHUNK == 0 for both GEMMs (4096, 384).
// ---------------------------------------------------------------------------
__global__ __launch_bounds__(128) void gemm_l2norm_wmma(
    const float* __restrict__ A, const float* __restrict__ W,
    const float* __restrict__ bias, float* __restrict__ out,
    int M, int K)
{
    __shared__ __align__(16) char smem[SMEM_BYTES];
    float* ldsB = (float*)smem;                   // W chunk, operand-pair order
    float* ldsA = (float*)(smem + LDSB_BYTES);    // TDM destination tile
    float* tile = (float*)smem;                   // epilogue alias: [4][16][72]
    float* rowScale = (float*)(smem + 4 * 16 * 72 * 4);

    const int tid    = threadIdx.x;
    const int wave   = tid >> 5;
    const int lane   = tid & 31;
    const int lanelo = lane & 15;
    const int kh2    = lane >> 4;                 // 0: K in {0,1}, 1: K in {2,3}
    const int kHalf  = kh2 << 1;

    const int rowBase0 = blockIdx.x * 64;
    const int rowBase  = rowBase0 + wave * 16;

    // Low 32 bits of a flat LDS pointer are the workgroup-relative LDS offset.
    const unsigned ldsAoff = (unsigned)(size_t)ldsA;

    v8f c0 = {}; v8f c1 = {}; v8f c2 = {}; v8f c3 = {};

    const int aRowOff = (wave * 16 + lanelo) * A_STRIDE + kHalf;
    const int bOff    = kh2 * B_STRIDE + 2 * lanelo;

    for (int kc = 0; kc < K; kc += KCHUNK) {
        // ---- TDM: async 2D tile load A[rowBase0:+64, kc:+KCHUNK] -> ldsA ----
        if (wave == 0) {
            unsigned long ga = (unsigned long)(const void*)(A + (long)rowBase0 * K + kc);
            v4u g0;
            g0[0] = 1u;                                            // count=1
            g0[1] = ldsAoff;                                       // lds_addr
            g0[2] = (unsigned)ga;                                  // global_addr[31:0]
            g0[3] = (unsigned)((ga >> 32) & 0x01FFFFFFu)           // global_addr[56:32]
                    | 0x80000000u;                                 // type=2 (image)
            const unsigned td0 = (unsigned)(K - kc);               // x extent from tile
            const unsigned td1 = (unsigned)(M - rowBase0);         // rows left (OOB->0)
            v8i g1;
            g1[0] = (int)((2u << 16)      // data_size = 4B
                        | (1u << 20)      // pad_enable
                        | (5u << 22)      // pad_interval: 64 dwords (= tile_dim0)
                        | (33u << 25));   // pad_amount: 34 dwords -> row stride 98
            g1[1] = (int)((td0 & 0xFFFFu) << 16);                  // tensor_dim0 lo
            g1[2] = (int)((td0 >> 16) | ((td1 & 0xFFFFu) << 16));  // dim0 hi | dim1 lo
            g1[3] = (int)(((td1 >> 16) & 0xFFFFu)
                        | ((unsigned)KCHUNK << 16));               // dim1 hi | tile_dim0
            g1[4] = 64;                                            // tile_dim1 = 64 rows
            g1[5] = K;                                             // tensor_dim0_stride lo
            g1[6] = 0;
            g1[7] = 0;
            v4i gz4 = {0, 0, 0, 0};
            v8i gz8 = {0, 0, 0, 0, 0, 0, 0, 0};
            __builtin_amdgcn_tensor_load_to_lds(g0, g1, gz4, gz4, gz8, 0);
        }

        // ---- stage W chunk into operand-pair order: {W[k][n], W[k+1][n]} ----
        for (int idx = tid; idx < KCHUNK * 64; idx += 128) {
            const int kr = idx >> 6, n = idx & 63;
            ldsB[(kr >> 1) * B_STRIDE + 2 * n + (kr & 1)] = W[(long)(kc + kr) * 64 + n];
        }

        if (wave == 0) __builtin_amdgcn_s_wait_tensorcnt(0);
        __syncthreads();

        #pragma unroll
        for (int k = 0; k < KCHUNK; k += 4) {
            // A operand: lane m holds A[m][kIdx..kIdx+1] — one ds_load_b64
            v2f a = *(const v2f*)&ldsA[aRowOff + k];
            // B operands: lane n holds B[kIdx][n],B[kIdx+1][n] — ds_load_b64 each
            const float* bp = &ldsB[(k >> 1) * B_STRIDE + bOff];
            v2f b0 = *(const v2f*)(bp +  0);
            v2f b1 = *(const v2f*)(bp + 32);
            v2f b2 = *(const v2f*)(bp + 64);
            v2f b3 = *(const v2f*)(bp + 96);
            c0 = __builtin_amdgcn_wmma_f32_16x16x4_f32(false, a, false, b0, (short)0, c0, false, false);
            c1 = __builtin_amdgcn_wmma_f32_16x16x4_f32(false, a, false, b1, (short)0, c1, false, false);
            c2 = __builtin_amdgcn_wmma_f32_16x16x4_f32(false, a, false, b2, (short)0, c2, false, false);
            c3 = __builtin_amdgcn_wmma_f32_16x16x4_f32(false, a, false, b3, (short)0, c3, false, false);
        }
        __syncthreads();  // also fences LDS before epilogue aliasing / next stage
    }

    // ---- epilogue: bias, row L2 norm, store (tile aliases staging LDS) ----
    const int rHalf = kh2 * 8;
    #pragma unroll
    for (int v = 0; v < 8; ++v) {
        const int r = (wave * 16 + rHalf + v) * 72;
        tile[r + lanelo +  0] = c0[v] + bias[lanelo +  0];
        tile[r + lanelo + 16] = c1[v] + bias[lanelo + 16];
        tile[r + lanelo + 32] = c2[v] + bias[lanelo + 32];
        tile[r + lanelo + 48] = c3[v] + bias[lanelo + 48];
    }
    __syncthreads();

    if (lane < 16) {
        float s = 0.f;
        #pragma unroll
        for (int n = 0; n < 64; ++n) {
            const float x = tile[(wave * 16 + lane) * 72 + n];
            s += x * x;
        }
        rowScale[wave * 16 + lane] = 1.0f / fmaxf(sqrtf(s), 1e-12f);
    }
    __syncthreads();

    #pragma unroll
    for (int r = 0; r < 16; ++r) {
        const int row = rowBase + r;
        if (row < M) {
            const float sc = rowScale[wave * 16 + r];
            out[(long)row * 64 + 2 * lane + 0] = tile[(wave * 16 + r) * 72 + 2 * lane + 0] * sc;
            out[(long)row * 64 + 2 * lane + 1] = tile[(wave * 16 + r) * 72 + 2 * lane + 1] * sc;
        }
    }
}

// ---------------------------------------------------------------------------
// out = ALPHA * in  (initializes the accumulation target for a layer)
// ---------------------------------------------------------------------------
__global__ __launch_bounds__(256) void scale_init(
    const float4* __restrict__ in, float4* __restrict__ out, int n4)
{
    int i = blockIdx.x * blockDim.x + threadIdx.x;
    if (i >= n4) return;
    float4 v = in[i];
    out[i] = make_float4(ALPHA_F * v.x, ALPHA_F * v.y, ALPHA_F * v.z, ALPHA_F * v.w);
}

// ---------------------------------------------------------------------------
// COO SpMM: one wave per edge, both modalities fused (indices read once).
// out[row] += val * x[col]; ego buffers (38 MB each) stay L2-resident, so
// the gathers and f32 atomics are L2 traffic on the 192 MB L2.
// ---------------------------------------------------------------------------
__global__ __launch_bounds__(256) void spmm_edges(
    const int* __restrict__ rows, const int* __restrict__ cols,
    const float* __restrict__ vals,
    const float* __restrict__ xi, const float* __restrict__ xt,
    float* __restrict__ oi, float* __restrict__ ot, int nnz)
{
    const int wave = (int)(((unsigned)blockIdx.x * 256u + threadIdx.x) >> 5);
    const int lane = threadIdx.x & 31;
    if (wave >= nnz) return;

    const int   r = rows[wave];
    const int   c = cols[wave];
    const float v = vals[wave];

    const long cb = (long)c * D_EMB + lane;
    const long rb = (long)r * D_EMB + lane;

    const float i0 = v * xi[cb];
    const float i1 = v * xi[cb + 32];
    const float t0 = v * xt[cb];
    const float t1 = v * xt[cb + 32];

    unsafeAtomicAdd(&oi[rb],      i0);
    unsafeAtomicAdd(&oi[rb + 32], i1);
    unsafeAtomicAdd(&ot[rb],      t0);
    unsafeAtomicAdd(&ot[rb + 32], t1);
}

// ---------------------------------------------------------------------------
// Output: row r of d_out = concat(ego_i[r], ego_t[r]); user rows then item
// rows are already in the right order in the ego tensors.
// ---------------------------------------------------------------------------
__global__ __launch_bounds__(256) void concat_out(
    const float4* __restrict__ ei, const float4* __restrict__ et,
    float4* __restrict__ out, int n4)
{
    int i = blockIdx.x * blockDim.x + threadIdx.x;
    if (i >= n4) return;
    int r  = i >> 5;          // 32 float4 per 128-wide output row
    int d4 = i & 31;
    out[i] = (d4 < 16) ? ei[(long)r * 16 + d4] : et[(long)r * 16 + (d4 - 16)];
}

// ---------------------------------------------------------------------------
extern "C" void kernel_launch(void* const* d_in, const int* in_sizes, int n_in,
                              void* d_out, int out_size, void* d_ws, size_t ws_size,
                              hipStream_t stream)
{
    const float* image_feats = (const float*)d_in[0];
    const float* text_feats  = (const float*)d_in[1];
    const float* image_pref  = (const float*)d_in[2];
    const float* text_pref   = (const float*)d_in[3];
    const float* W_img       = (const float*)d_in[4];
    const float* b_img       = (const float*)d_in[5];
    const float* W_txt       = (const float*)d_in[6];
    const float* b_txt       = (const float*)d_in[7];
    const float* adj_vals    = (const float*)d_in[8];
    const int*   adj_rows    = (const int*)d_in[9];
    const int*   adj_cols    = (const int*)d_in[10];
    const int    nnz         = in_sizes[9];

    const size_t S = (size_t)N_TOT * D_EMB;       // floats per modality buffer
    float* bufA = (float*)d_ws;                   // [2S]: ego_i then ego_t
    float* bufB = bufA + 2 * S;                   // [2S]: ping-pong target

    // ego[:N_USERS] = prefs (device-to-device async copies are capture-safe)
    (void)hipMemcpyAsync(bufA,     image_pref, (size_t)N_USERS * D_EMB * sizeof(float),
                         hipMemcpyDeviceToDevice, stream);
    (void)hipMemcpyAsync(bufA + S, text_pref,  (size_t)N_USERS * D_EMB * sizeof(float),
                         hipMemcpyDeviceToDevice, stream);

    // ego[N_USERS:] = l2norm(feats @ W + b)
    const int gemmBlocks = (N_ITEMS + 63) / 64;
    gemm_l2norm_wmma<<<gemmBlocks, 128, 0, stream>>>(
        image_feats, W_img, b_img, bufA + (size_t)N_USERS * D_EMB, N_ITEMS, IMG_K);
    gemm_l2norm_wmma<<<gemmBlocks, 128, 0, stream>>>(
        text_feats, W_txt, b_txt, bufA + S + (size_t)N_USERS * D_EMB, N_ITEMS, TXT_K);

    // 3 propagation layers: B = ALPHA*A, then B += adj @ A (both modalities)
    float* Acur = bufA;
    float* Bcur = bufB;
    const int n4scale = (int)((2 * S) / 4);
    for (int layer = 0; layer < 3; ++layer) {
        scale_init<<<(n4scale + 255) / 256, 256, 0, stream>>>(
            (const float4*)Acur, (float4*)Bcur, n4scale);
        spmm_edges<<<(nnz + 7) / 8, 256, 0, stream>>>(
            adj_rows, adj_cols, adj_vals,
            Acur, Acur + S, Bcur, Bcur + S, nnz);
        float* t = Acur; Acur = Bcur; Bcur = t;
    }

    const int n4out = out_size / 4;
    concat_out<<<(n4out + 255) / 256, 256, 0, stream>>>(
        (const float4*)Acur, (const float4*)(Acur + S), (float4*)d_out, n4out);
}